// subplanar_pdd_74569222193580
// MI455X (gfx1250) — compile-verified
//
#include <hip/hip_runtime.h>

typedef __attribute__((ext_vector_type(2)))  float     v2f;
typedef __attribute__((ext_vector_type(4)))  float     v4f;
typedef __attribute__((ext_vector_type(8)))  float     v8f;

#define Cch 16
#define Ssz 64
#define S3  262144      // 64^3
#define Gg  29
#define G3  24389       // 29^3
#define Dd  15
#define Kk  675         // 15*15*3
#define NTOT 16462575   // G3*Kk

// ---------------------------------------------------------------------------
// K0: transpose [C, S^3] -> [S^3, C]  (channel-last so one tap = 64B run)
// ---------------------------------------------------------------------------
__global__ void k_transpose_cl(const float* __restrict__ src, float* __restrict__ dst) {
    __shared__ float tile[64][17];
    const int vbase = blockIdx.x * 64;
    const int t = threadIdx.x;            // 256 threads
#pragma unroll
    for (int i = 0; i < 4; ++i) {
        int e  = t + i * 256;             // 0..1023
        int ch = e >> 6;                  // 0..15
        int vo = e & 63;
        tile[vo][ch] = src[(size_t)ch * S3 + vbase + vo];
    }
    __syncthreads();
#pragma unroll
    for (int i = 0; i < 4; ++i) {
        int e  = t + i * 256;
        int vo = e >> 4;
        int c  = e & 15;
        dst[((size_t)(vbase + vo) << 4) + c] = tile[vo][c];
    }
}

// ---------------------------------------------------------------------------
// K1: cost volume + pdd = a1 + a0 * sum_c (fix_c - mov_c)^2
// one block per grid point g; fix computed once into LDS; threads sweep k.
// ---------------------------------------------------------------------------
__global__ void k_cost_pdd(const float* __restrict__ f00t, const float* __restrict__ f50t,
                           const float* __restrict__ grid_xyz, const float* __restrict__ sh2d,
                           const float* __restrict__ alpha, float* __restrict__ pdd) {
    const int g   = blockIdx.x;
    const int tid = threadIdx.x;
    __shared__ float fixv[16];

    const float gx = grid_xyz[g * 3 + 0];
    const float gy = grid_xyz[g * 3 + 1];
    const float gz = grid_xyz[g * 3 + 2];

    if (tid < 16) {
        // zeros-mode trilinear sample of feat00 channel `tid` at grid point
        float fx = ((gx + 1.f) * 64.f - 1.f) * 0.5f;
        float fy = ((gy + 1.f) * 64.f - 1.f) * 0.5f;
        float fz = ((gz + 1.f) * 64.f - 1.f) * 0.5f;
        float x0f = floorf(fx), y0f = floorf(fy), z0f = floorf(fz);
        float wx = fx - x0f, wy = fy - y0f, wz = fz - z0f;
        float acc = 0.f;
#pragma unroll
        for (int dz = 0; dz < 2; ++dz)
#pragma unroll
            for (int dy = 0; dy < 2; ++dy)
#pragma unroll
                for (int dx = 0; dx < 2; ++dx) {
                    float xf = x0f + dx, yf = y0f + dy, zf = z0f + dz;
                    bool inb = (xf >= 0.f) && (xf <= 63.f) && (yf >= 0.f) && (yf <= 63.f)
                            && (zf >= 0.f) && (zf <= 63.f);
                    float w = (dx ? wx : 1.f - wx) * (dy ? wy : 1.f - wy) * (dz ? wz : 1.f - wz);
                    if (inb) {
                        int xi = (int)xf, yi = (int)yf, zi = (int)zf;
                        acc += w * f00t[((size_t)((zi * 64 + yi) * 64 + xi) << 4) + tid];
                    }
                }
        fixv[tid] = acc;
    }
    __syncthreads();

    const float a0 = alpha[0], a1 = alpha[1];
    const v4f f0 = *(const v4f*)(fixv + 0);
    const v4f f1 = *(const v4f*)(fixv + 4);
    const v4f f2 = *(const v4f*)(fixv + 8);
    const v4f f3 = *(const v4f*)(fixv + 12);

    for (int k = tid; k < Kk; k += blockDim.x) {
        float x = gx + sh2d[k * 3 + 0];
        float y = gy + sh2d[k * 3 + 1];
        float z = gz + sh2d[k * 3 + 2];
        // border-mode trilinear sample of feat50 (all 16 channels)
        float fx = fminf(fmaxf(((x + 1.f) * 64.f - 1.f) * 0.5f, 0.f), 63.f);
        float fy = fminf(fmaxf(((y + 1.f) * 64.f - 1.f) * 0.5f, 0.f), 63.f);
        float fz = fminf(fmaxf(((z + 1.f) * 64.f - 1.f) * 0.5f, 0.f), 63.f);
        float x0f = floorf(fx), y0f = floorf(fy), z0f = floorf(fz);
        int x0 = (int)x0f, y0 = (int)y0f, z0 = (int)z0f;
        int xi2[2] = { x0, min(x0 + 1, 63) };
        int yi2[2] = { y0, min(y0 + 1, 63) };
        int zi2[2] = { z0, min(z0 + 1, 63) };
        float wxv[2] = { 1.f - (fx - x0f), fx - x0f };
        float wyv[2] = { 1.f - (fy - y0f), fy - y0f };
        float wzv[2] = { 1.f - (fz - z0f), fz - z0f };

        v4f m0 = {0,0,0,0}, m1 = {0,0,0,0}, m2 = {0,0,0,0}, m3 = {0,0,0,0};
        __builtin_prefetch(f50t + ((size_t)((zi2[0] * 64 + yi2[0]) * 64 + xi2[0]) << 4), 0, 1);
#pragma unroll
        for (int dz = 0; dz < 2; ++dz)
#pragma unroll
            for (int dy = 0; dy < 2; ++dy)
#pragma unroll
                for (int dx = 0; dx < 2; ++dx) {
                    float w = wzv[dz] * wyv[dy] * wxv[dx];
                    const v4f* p = (const v4f*)(f50t +
                        ((size_t)((zi2[dz] * 64 + yi2[dy]) * 64 + xi2[dx]) << 4));
                    m0 += w * p[0]; m1 += w * p[1]; m2 += w * p[2]; m3 += w * p[3];
                }
        v4f d0 = f0 - m0, d1 = f1 - m1, d2 = f2 - m2, d3 = f3 - m3;
        v4f s = d0 * d0 + d1 * d1 + d2 * d2 + d3 * d3;
        float cost = s.x + s.y + s.z + s.w;
        pdd[(size_t)g * Kk + k] = a1 + a0 * cost;
    }
}

// ---------------------------------------------------------------------------
// K2: replication-pad(2) -> 3x3 min -> 3x3 avg over the (D,D) disp dims.
// ---------------------------------------------------------------------------
__global__ void k_min_then_avg(const float* __restrict__ in, float* __restrict__ out) {
    size_t idx = (size_t)blockIdx.x * 256 + threadIdx.x;
    if (idx >= (size_t)NTOT) return;
    int pl = (int)(idx % 3);
    size_t t = idx / 3;
    int j = (int)(t % Dd); t /= Dd;
    int i = (int)(t % Dd);
    size_t g = t / Dd;
    const float* base = in + g * Kk + pl;

    float patch[5][5];
#pragma unroll
    for (int u = 0; u < 5; ++u) {
        int iu = min(max(i + u - 2, 0), Dd - 1);
#pragma unroll
        for (int v = 0; v < 5; ++v) {
            int jv = min(max(j + v - 2, 0), Dd - 1);
            patch[u][v] = base[(size_t)(iu * Dd + jv) * 3];
        }
    }
    float s = 0.f;
#pragma unroll
    for (int a = 0; a < 3; ++a)
#pragma unroll
        for (int b = 0; b < 3; ++b) {
            float m = patch[a][b];
#pragma unroll
            for (int c = 0; c < 3; ++c)
#pragma unroll
                for (int d = 0; d < 3; ++d) m = fminf(m, patch[a + c][b + d]);
            s += m;
        }
    out[idx] = s * (1.f / 9.f);
}

// ---------------------------------------------------------------------------
// K3: grid smooth: two 3x3 box avgs over (g0,g1) with edge pad 2
//     == single separable 5-tap [1,2,3,2,1]/9 per axis with edge clamp.
// ---------------------------------------------------------------------------
__global__ void k_grid_smooth(const float* __restrict__ in, float* __restrict__ out) {
    size_t idx = (size_t)blockIdx.x * 256 + threadIdx.x;
    if (idx >= (size_t)NTOT) return;
    int k = (int)(idx % Kk);
    size_t r = idx / Kk;
    int g2 = (int)(r % Gg); r /= Gg;
    int g1 = (int)(r % Gg);
    int g0 = (int)(r / Gg);
    const float w5[5] = {1.f, 2.f, 3.f, 2.f, 1.f};
    float s = 0.f;
#pragma unroll
    for (int u = 0; u < 5; ++u) {
        int iu = min(max(g0 + u - 2, 0), Gg - 1);
#pragma unroll
        for (int v = 0; v < 5; ++v) {
            int jv = min(max(g1 + v - 2, 0), Gg - 1);
            s += w5[u] * w5[v] * in[((size_t)((iu * Gg + jv) * Gg + g2)) * Kk + k];
        }
    }
    out[idx] = s * (1.f / 81.f);
}

// ---------------------------------------------------------------------------
// K4: cost2 = a4 + a2*pdd + a3*cost_avg1, fused with min_then_avg.
// ---------------------------------------------------------------------------
__global__ void k_cost2_minavg(const float* __restrict__ P, const float* __restrict__ A1,
                               const float* __restrict__ alpha, float* __restrict__ out) {
    size_t idx = (size_t)blockIdx.x * 256 + threadIdx.x;
    if (idx >= (size_t)NTOT) return;
    const float a2 = alpha[2], a3 = alpha[3], a4 = alpha[4];
    int pl = (int)(idx % 3);
    size_t t = idx / 3;
    int j = (int)(t % Dd); t /= Dd;
    int i = (int)(t % Dd);
    size_t g = t / Dd;
    const float* bp = P  + g * Kk + pl;
    const float* ba = A1 + g * Kk + pl;

    float patch[5][5];
#pragma unroll
    for (int u = 0; u < 5; ++u) {
        int iu = min(max(i + u - 2, 0), Dd - 1);
#pragma unroll
        for (int v = 0; v < 5; ++v) {
            int jv = min(max(j + v - 2, 0), Dd - 1);
            size_t o = (size_t)(iu * Dd + jv) * 3;
            patch[u][v] = a4 + a2 * bp[o] + a3 * ba[o];
        }
    }
    float s = 0.f;
#pragma unroll
    for (int a = 0; a < 3; ++a)
#pragma unroll
        for (int b = 0; b < 3; ++b) {
            float m = patch[a][b];
#pragma unroll
            for (int c = 0; c < 3; ++c)
#pragma unroll
                for (int d = 0; d < 3; ++d) m = fminf(m, patch[a + c][b + d]);
            s += m;
        }
    out[idx] = s * (1.f / 9.f);
}

// ---------------------------------------------------------------------------
// K6: softmax over the grid axis (24389) for each of the 675 (p,c) columns.
// ---------------------------------------------------------------------------
__global__ void k_softmax_col(const float* __restrict__ x, const float* __restrict__ alpha,
                              float* __restrict__ out) {
    const int col = blockIdx.x;          // 0..674
    const int tid = threadIdx.x;         // 256
    const float a5 = alpha[5];
    __shared__ float red[256];

    float m = -3.402823466e+38f;
    for (int g = tid; g < G3; g += 256) m = fmaxf(m, -a5 * x[(size_t)g * Kk + col]);
    red[tid] = m; __syncthreads();
    for (int off = 128; off > 0; off >>= 1) {
        if (tid < off) red[tid] = fmaxf(red[tid], red[tid + off]);
        __syncthreads();
    }
    m = red[0]; __syncthreads();

    float s = 0.f;
    for (int g = tid; g < G3; g += 256) s += __expf(-a5 * x[(size_t)g * Kk + col] - m);
    red[tid] = s; __syncthreads();
    for (int off = 128; off > 0; off >>= 1) {
        if (tid < off) red[tid] += red[tid + off];
        __syncthreads();
    }
    const float inv = 1.f / red[0];

    for (int g = tid; g < G3; g += 256)
        out[(size_t)g * Kk + col] = __expf(-a5 * x[(size_t)g * Kk + col] - m) * inv;
}

// ---------------------------------------------------------------------------
// K7: pred_xyz = 0.5 * cost_soft[G3 x 675] @ shift[675 x 3]  via
// V_WMMA_F32_16X16X4_F32. One wave per 32-row tile (two accumulator chains),
// fully branchless inner loop: row pointers clamped once, B columns n>=3
// zeroed with an arithmetic mask, single masked tail for k=672..674.
// ---------------------------------------------------------------------------
__global__ void k_pred_wmma(const float* __restrict__ cs, const float* __restrict__ sh,
                            float* __restrict__ pred) {
    const int lane = threadIdx.x & 31;
    const int half = lane >> 4;               // A/B K sub-offset: 0 -> K+0/1, 1 -> K+2/3
    const int l    = lane & 15;
    const int base = blockIdx.x * 32;

    // clamped A row pointers (OOB rows read a valid duplicate row; never stored)
    const float* arow0 = cs + (size_t)min(base + l,      G3 - 1) * Kk;
    const float* arow1 = cs + (size_t)min(base + 16 + l, G3 - 1) * Kk;
    // B: value 0.5*sh[k*3+n] for n<3 else 0 (clamped column, arithmetic mask)
    const int   lc    = min(l, 2);
    const float bmask = (l < 3) ? 0.5f : 0.0f;

    v8f acc0 = {}, acc1 = {};
    const int koff = half * 2;

    for (int kk = 0; kk < 672; kk += 8) {
        const int ka = kk + koff;             // chunk 0: k = ka, ka+1
        const int kb = kk + 4 + koff;         // chunk 1: k = kb, kb+1
        v2f aA0, aA1, aB0, aB1, bA, bB;
        aA0.x = arow0[ka];     aA0.y = arow0[ka + 1];
        aA1.x = arow1[ka];     aA1.y = arow1[ka + 1];
        aB0.x = arow0[kb];     aB0.y = arow0[kb + 1];
        aB1.x = arow1[kb];     aB1.y = arow1[kb + 1];
        bA.x  = sh[ka * 3 + lc] * bmask;  bA.y = sh[(ka + 1) * 3 + lc] * bmask;
        bB.x  = sh[kb * 3 + lc] * bmask;  bB.y = sh[(kb + 1) * 3 + lc] * bmask;
        acc0 = __builtin_amdgcn_wmma_f32_16x16x4_f32(false, aA0, false, bA, (short)0, acc0,
                                                     false, false);
        acc1 = __builtin_amdgcn_wmma_f32_16x16x4_f32(false, aA1, false, bA, (short)0, acc1,
                                                     false, false);
        acc0 = __builtin_amdgcn_wmma_f32_16x16x4_f32(false, aB0, false, bB, (short)0, acc0,
                                                     false, false);
        acc1 = __builtin_amdgcn_wmma_f32_16x16x4_f32(false, aB1, false, bB, (short)0, acc1,
                                                     false, false);
    }
    {   // tail: k = 672, 673, 674 (pad k=675 with zero via mask, no branches)
        const int   ka  = 672 + koff;
        const int   ka1 = min(ka + 1, Kk - 1);
        const float m1  = (ka + 1 < Kk) ? 1.f : 0.f;
        v2f a0, a1, b;
        a0.x = arow0[ka];  a0.y = arow0[ka1] * m1;
        a1.x = arow1[ka];  a1.y = arow1[ka1] * m1;
        b.x  = sh[ka * 3 + lc] * bmask;
        b.y  = sh[ka1 * 3 + lc] * bmask * m1;
        acc0 = __builtin_amdgcn_wmma_f32_16x16x4_f32(false, a0, false, b, (short)0, acc0,
                                                     false, false);
        acc1 = __builtin_amdgcn_wmma_f32_16x16x4_f32(false, a1, false, b, (short)0, acc1,
                                                     false, false);
    }
    if (l < 3) {
#pragma unroll
        for (int r = 0; r < 8; ++r) {
            int row0 = base + r + half * 8;
            int row1 = base + 16 + r + half * 8;
            if (row0 < G3) pred[row0 * 3 + l] = acc0[r];
            if (row1 < G3) pred[row1 * 3 + l] = acc1[r];
        }
    }
}

// ---------------------------------------------------------------------------
extern "C" void kernel_launch(void* const* d_in, const int* in_sizes, int n_in,
                              void* d_out, int out_size, void* d_ws, size_t ws_size,
                              hipStream_t stream) {
    (void)in_sizes; (void)n_in; (void)out_size; (void)ws_size;
    const float* feat00   = (const float*)d_in[0];
    const float* feat50   = (const float*)d_in[1];
    /* d_in[2] = shift_2d_min: pure broadcast of shift_2d -> unused */
    const float* grid_xyz = (const float*)d_in[3];
    const float* shift2d  = (const float*)d_in[4];
    const float* alpha    = (const float*)d_in[5];

    const size_t S3C = (size_t)S3 * Cch;        // 4,194,304
    float* f00t = (float*)d_ws;
    float* f50t = f00t + S3C;
    float* P    = f50t + S3C;                   // pdd  [G3*Kk]
    float* T1   = P  + (size_t)NTOT;
    float* T2   = T1 + (size_t)NTOT;

    float* out_soft = (float*)d_out;                      // [G3*675]
    float* out_pred = out_soft + (size_t)NTOT;            // [G3*3]
    float* out_avg2 = out_pred + (size_t)G3 * 3;          // [G3*675]

    const int nb = (NTOT + 255) / 256;

    k_transpose_cl<<<S3 / 64, 256, 0, stream>>>(feat00, f00t);
    k_transpose_cl<<<S3 / 64, 256, 0, stream>>>(feat50, f50t);
    k_cost_pdd<<<G3, 256, 0, stream>>>(f00t, f50t, grid_xyz, shift2d, alpha, P);
    k_min_then_avg<<<nb, 256, 0, stream>>>(P, T1);             // cost1
    k_grid_smooth<<<nb, 256, 0, stream>>>(T1, T2);             // cost_avg1
    k_cost2_minavg<<<nb, 256, 0, stream>>>(P, T2, alpha, T1);  // cost2 (post min/avg)
    k_grid_smooth<<<nb, 256, 0, stream>>>(T1, out_avg2);       // cost_avg2 -> output 3
    k_softmax_col<<<Kk, 256, 0, stream>>>(out_avg2, alpha, out_soft);      // output 1
    k_pred_wmma<<<(G3 + 31) / 32, 32, 0, stream>>>(out_soft, shift2d, out_pred); // output 2
}